// PointNetPP_80272938762727
// MI455X (gfx1250) — compile-verified
//
#include <hip/hip_runtime.h>
#include <cfloat>

// ============================================================================
// PointNet++ (SA x4 + FP x4) forward for MI455X (gfx1250, wave32).
// MLP GEMMs run on V_WMMA_F32_16X16X4_F32 (full f32 precision; ~75 GFLOP of
// matmul is trivial vs. the WMMA/HBM ceilings, so f32 matrix ops are free and
// reference-exact). SA edge-MLPs are fully fused (gather -> 3-layer MLP ->
// masked max): edge matrix + hidden activations live in LDS (<=100KB of the
// 320KB/WGP). All weights are pre-padded & K-pair-packed so every WMMA inner
// loop is unguarded b64 loads + v_wmma (no lane-divergent exec masking).
// ============================================================================

typedef float v2f __attribute__((ext_vector_type(2)));
typedef float v8f __attribute__((ext_vector_type(8)));

__device__ __forceinline__ v8f wmma16x16x4(v2f a, v2f b, v8f c) {
  // D = A(16x4) * B(4x16) + C(16x16), f32, wave32
  return __builtin_amdgcn_wmma_f32_16x16x4_f32(false, a, false, b, (short)0, c,
                                               false, false);
}

#define V8F_ZERO {0.f,0.f,0.f,0.f,0.f,0.f,0.f,0.f}

// ---------------------------------------------------------------------------
// Weight prep: zero-pad K to multiple of 4 and pack K-pairs so a B-fragment
// is one aligned b64 load:  Wp[((k>>1)*cout + j)*2 + (k&1)] = W[k][j] (or 0).
// ---------------------------------------------------------------------------
__global__ void pad_pack_kernel(const float* __restrict__ W, int cin, int cout,
                                float* __restrict__ Wp, int total) {
  int g = blockIdx.x * blockDim.x + threadIdx.x;
  if (g >= total) return;
  const int cout2 = cout << 1;
  const int p = g / cout2;
  const int rem = g % cout2;
  const int j = rem >> 1;
  const int k = (p << 1) + (rem & 1);
  Wp[g] = (k < cin) ? W[(size_t)k * cout + j] : 0.f;
}

// ---------------------------------------------------------------------------
// Farthest point sampling: one block per cloud, dist table in LDS.
// Matches jnp.argmax tie-break (lowest index).
// ---------------------------------------------------------------------------
__global__ void fps_kernel(const float* __restrict__ pos, int n, int m,
                           int* __restrict__ idx_out) {
  extern __shared__ float smem[];
  float* dist = smem;                       // n
  float* rbest = dist + n;                  // blockDim.x
  int* ribest = (int*)(rbest + blockDim.x); // blockDim.x
  const int b = blockIdx.x;
  const float* P = pos + (size_t)b * n * 3;
  const int tid = threadIdx.x, bs = blockDim.x;
  const float x0 = P[0], y0 = P[1], z0 = P[2];
  for (int i = tid; i < n; i += bs) {
    float dx = P[3 * i] - x0, dy = P[3 * i + 1] - y0, dz = P[3 * i + 2] - z0;
    dist[i] = dx * dx + dy * dy + dz * dz;
  }
  if (tid == 0) idx_out[(size_t)b * m] = 0;
  __syncthreads();
  for (int j = 1; j < m; ++j) {
    float best = -1.f; int bi = n;
    for (int i = tid; i < n; i += bs) {
      float d = dist[i];
      if (d > best) { best = d; bi = i; }
    }
    rbest[tid] = best; ribest[tid] = bi;
    __syncthreads();
    for (int s = bs >> 1; s > 0; s >>= 1) {
      if (tid < s) {
        float o = rbest[tid + s];
        if (o > rbest[tid] || (o == rbest[tid] && ribest[tid + s] < ribest[tid])) {
          rbest[tid] = o; ribest[tid] = ribest[tid + s];
        }
      }
      __syncthreads();
    }
    const int sel = ribest[0];
    if (tid == 0) idx_out[(size_t)b * m + j] = sel;
    const float sx = P[3 * sel], sy = P[3 * sel + 1], sz = P[3 * sel + 2];
    __syncthreads();
    for (int i = tid; i < n; i += bs) {
      float dx = P[3 * i] - sx, dy = P[3 * i + 1] - sy, dz = P[3 * i + 2] - sz;
      float nd = dx * dx + dy * dy + dz * dz;
      if (nd < dist[i]) dist[i] = nd;
    }
    __syncthreads();
  }
}

// ---------------------------------------------------------------------------
// Gather query positions pos_q[b,q] = pos_src[b, idx[b,q]]
// ---------------------------------------------------------------------------
__global__ void gather_pos_kernel(const float* __restrict__ pos_src, int n,
                                  const int* __restrict__ idx, int m,
                                  float* __restrict__ pos_q, int total) {
  int t = blockIdx.x * blockDim.x + threadIdx.x;
  if (t >= total) return;
  int b = t / m;
  int s = idx[t];
  const float* ps = pos_src + ((size_t)b * n + s) * 3;
  pos_q[(size_t)t * 3 + 0] = ps[0];
  pos_q[(size_t)t * 3 + 1] = ps[1];
  pos_q[(size_t)t * 3 + 2] = ps[2];
}

// ---------------------------------------------------------------------------
// Radius neighbors: first K (=32) in-radius sources in ascending index order
// (matches stable argsort on the within-flag). Thread per query.
// ---------------------------------------------------------------------------
__global__ void radius_kernel(const float* __restrict__ pos_src, int n,
                              const float* __restrict__ pos_q, int m, float r2,
                              int* __restrict__ nbr, int* __restrict__ cnt,
                              int total) {
  int t = blockIdx.x * blockDim.x + threadIdx.x;
  if (t >= total) return;
  int b = t / m;
  const float* P = pos_src + (size_t)b * n * 3;
  const float qx = pos_q[(size_t)t * 3 + 0];
  const float qy = pos_q[(size_t)t * 3 + 1];
  const float qz = pos_q[(size_t)t * 3 + 2];
  int* out = nbr + (size_t)t * 32;
  int c = 0;
  for (int i = 0; i < n; ++i) {
    float dx = P[3 * i] - qx, dy = P[3 * i + 1] - qy, dz = P[3 * i + 2] - qz;
    if (dx * dx + dy * dy + dz * dz <= r2) {
      out[c++] = i;
      if (c == 32) break;
    }
  }
  for (int j = c; j < 32; ++j) out[j] = 0;
  cnt[t] = c;
}

// ---------------------------------------------------------------------------
// Fused SA PointNetConv: per query, 32 neighbor rows through
// relu(W1)+relu(W2)+W3 edge MLP (WMMA f32 16x16x4), masked max-aggregate.
// Block = 128 threads (4 waves). Edge matrix is staged into LDS once
// (coalesced, zero-padded), so all three layers' inner loops are pure
// ds_load_b64 / global_load_b64 + v_wmma with uniform control flow.
// Weights are pre-packed by pad_pack_kernel.
// ---------------------------------------------------------------------------
__global__ void sa_conv_kernel(
    const float* __restrict__ x_src, int c_src, int n_src,
    const float* __restrict__ pos_src, const float* __restrict__ pos_q,
    const int* __restrict__ nbr, const int* __restrict__ cnt,
    const float* __restrict__ Wp1, const float* __restrict__ B1, int c1,
    const float* __restrict__ Wp2, const float* __restrict__ B2, int c2,
    const float* __restrict__ Wp3, const float* __restrict__ B3, int c3,
    float* __restrict__ x_out, int m) {
  extern __shared__ float smem[];
  const int lda1 = (c_src + 6) & ~3;  // round4(c_src + 3)
  float* a_s = smem;                   // 32*lda1
  float* h1s = a_s + 32 * lda1;        // 32*c1
  float* h2s = h1s + 32 * c1;          // 32*c2
  float* rel_s = h2s + 32 * c2;        // 32*3
  int* nbr_s = (int*)(rel_s + 96);     // 32

  const int t = blockIdx.x;            // b*m + q
  const int b = t / m;
  const int tid = threadIdx.x;
  const int lane = tid & 31;
  const int wv = tid >> 5;
  const int row = lane & 15;
  const int kh = (lane >> 4) << 1;     // K offset {0,2} by lane half
  const int hsel = (lane >> 4) << 3;   // C-frag row offset {0,8}
  const int myCnt = cnt[t];
  const float* x_b = x_src + (size_t)b * n_src * c_src;

  if (tid < 32) {
    int nn = 0; float rx = 0.f, ry = 0.f, rz = 0.f;
    if (tid < myCnt) {
      nn = nbr[(size_t)t * 32 + tid];
      const float* ps = pos_src + ((size_t)b * n_src + nn) * 3;
      const float* pq = pos_q + (size_t)t * 3;
      rx = ps[0] - pq[0]; ry = ps[1] - pq[1]; rz = ps[2] - pq[2];
    }
    nbr_s[tid] = nn;
    rel_s[tid * 3 + 0] = rx; rel_s[tid * 3 + 1] = ry; rel_s[tid * 3 + 2] = rz;
  }
  __syncthreads();

  // ---- stage edge matrix [32 x lda1] into LDS (coalesced along K) ----
  for (int kk = 0; kk < 32; ++kk) {
    const size_t src = (size_t)nbr_s[kk] * c_src;
    for (int k = tid; k < lda1; k += 128) {
      float v;
      if (k < c_src) v = x_b[src + k];
      else {
        int kr = k - c_src;
        v = (kr < 3) ? rel_s[kk * 3 + kr] : 0.f;
      }
      a_s[kk * lda1 + k] = v;
    }
  }
  __syncthreads();

  // ---- layer 1: a_s @ W1 -> h1s (relu) ----
  for (int jt = wv; jt < (c1 >> 4); jt += 4) {
    const int j = (jt << 4) + row;
    v8f acc0 = V8F_ZERO, acc1 = V8F_ZERO;
    for (int k = kh; k < lda1; k += 4) {
      v2f a0 = *(const v2f*)(a_s + row * lda1 + k);
      v2f a1 = *(const v2f*)(a_s + (16 + row) * lda1 + k);
      v2f bb = *(const v2f*)(Wp1 + (((size_t)(k >> 1) * c1 + j) << 1));
      acc0 = wmma16x16x4(a0, bb, acc0);
      acc1 = wmma16x16x4(a1, bb, acc1);
    }
    const float bias = B1[j];
#pragma unroll
    for (int v = 0; v < 8; ++v) {
      int kk = v + hsel;
      h1s[kk * c1 + j] = fmaxf(acc0[v] + bias, 0.f);
      h1s[(16 + kk) * c1 + j] = fmaxf(acc1[v] + bias, 0.f);
    }
  }
  __syncthreads();

  // ---- layer 2: h1s @ W2 -> h2s (relu) ----
  for (int jt = wv; jt < (c2 >> 4); jt += 4) {
    const int j = (jt << 4) + row;
    v8f acc0 = V8F_ZERO, acc1 = V8F_ZERO;
    for (int k = kh; k < c1; k += 4) {
      v2f a0 = *(const v2f*)(h1s + row * c1 + k);
      v2f a1 = *(const v2f*)(h1s + (16 + row) * c1 + k);
      v2f bb = *(const v2f*)(Wp2 + (((size_t)(k >> 1) * c2 + j) << 1));
      acc0 = wmma16x16x4(a0, bb, acc0);
      acc1 = wmma16x16x4(a1, bb, acc1);
    }
    const float bias = B2[j];
#pragma unroll
    for (int v = 0; v < 8; ++v) {
      int kk = v + hsel;
      h2s[kk * c2 + j] = fmaxf(acc0[v] + bias, 0.f);
      h2s[(16 + kk) * c2 + j] = fmaxf(acc1[v] + bias, 0.f);
    }
  }
  __syncthreads();

  // ---- layer 3 + masked max aggregation over 32 neighbor rows ----
  for (int jt = wv; jt < (c3 >> 4); jt += 4) {
    const int j = (jt << 4) + row;
    v8f acc0 = V8F_ZERO, acc1 = V8F_ZERO;
    for (int k = kh; k < c2; k += 4) {
      v2f a0 = *(const v2f*)(h2s + row * c2 + k);
      v2f a1 = *(const v2f*)(h2s + (16 + row) * c2 + k);
      v2f bb = *(const v2f*)(Wp3 + (((size_t)(k >> 1) * c3 + j) << 1));
      acc0 = wmma16x16x4(a0, bb, acc0);
      acc1 = wmma16x16x4(a1, bb, acc1);
    }
    const float bias = B3[j];
    float mx = -FLT_MAX;
#pragma unroll
    for (int v = 0; v < 8; ++v) {
      int kk = v + hsel;
      if (kk < myCnt) mx = fmaxf(mx, acc0[v] + bias);
      if (16 + kk < myCnt) mx = fmaxf(mx, acc1[v] + bias);
    }
    mx = fmaxf(mx, __shfl_xor(mx, 16, 32));
    if (lane < 16)
      x_out[(size_t)t * c3 + (jt << 4) + lane] = (myCnt > 0) ? mx : 0.f;
  }
}

// ---------------------------------------------------------------------------
// Generic WMMA GEMM: C[R,cout] = act(A[R,lda] @ W + bias), W pre-packed
// (pad_pack_kernel with ldk == lda). One wave per 16-row x 64-col tile.
// A must be zero-padded to lda (multiple of 4). All loads unguarded b64.
// ---------------------------------------------------------------------------
__global__ void gemm_kernel(const float* __restrict__ A, int lda,
                            const float* __restrict__ Wp, int cout,
                            const float* __restrict__ bias,
                            float* __restrict__ C, int ldc, int relu) {
  const int lane = threadIdx.x & 31;
  const int row = lane & 15;
  const int kh = (lane >> 4) << 1;
  const size_t r0 = (size_t)blockIdx.x << 4;
  const float* Arow = A + (r0 + row) * lda;
  for (int jt = 0; jt < 4; ++jt) {
    const int j0 = (((blockIdx.y << 2) + jt) << 4);
    if (j0 >= cout) break;
    const int j = j0 + row;
    v8f acc = V8F_ZERO;
    for (int k = kh; k < lda; k += 4) {
      __builtin_prefetch(Arow + k + 64, 0, 0);  // global_prefetch_b8
      v2f a = *(const v2f*)(Arow + k);
      v2f bb = *(const v2f*)(Wp + (((size_t)(k >> 1) * cout + j) << 1));
      acc = wmma16x16x4(a, bb, acc);
    }
    const float bs = bias[j];
#pragma unroll
    for (int v = 0; v < 8; ++v) {
      float val = acc[v] + bs;
      if (relu) val = fmaxf(val, 0.f);
      C[(r0 + v + ((lane >> 4) << 3)) * ldc + j] = val;
    }
  }
}

// ---------------------------------------------------------------------------
// kNN (k<=3) indices + 1/d^2 weights, thread per target. Stable ties.
// ---------------------------------------------------------------------------
__global__ void knn_kernel(const float* __restrict__ pos_src, int ns,
                           const float* __restrict__ pos_tgt, int nt, int k,
                           int* __restrict__ idx, float* __restrict__ w,
                           int total) {
  int t = blockIdx.x * blockDim.x + threadIdx.x;
  if (t >= total) return;
  int b = t / nt;
  const float* P = pos_src + (size_t)b * ns * 3;
  const float qx = pos_tgt[(size_t)t * 3 + 0];
  const float qy = pos_tgt[(size_t)t * 3 + 1];
  const float qz = pos_tgt[(size_t)t * 3 + 2];
  float d[3] = {FLT_MAX, FLT_MAX, FLT_MAX};
  int id[3] = {0, 0, 0};
  for (int i = 0; i < ns; ++i) {
    float dx = P[3 * i] - qx, dy = P[3 * i + 1] - qy, dz = P[3 * i + 2] - qz;
    float d2 = dx * dx + dy * dy + dz * dz;
    if (d2 < d[0]) {
      d[2] = d[1]; id[2] = id[1];
      d[1] = d[0]; id[1] = id[0];
      d[0] = d2;   id[0] = i;
    } else if (d2 < d[1]) {
      d[2] = d[1]; id[2] = id[1];
      d[1] = d2;   id[1] = i;
    } else if (d2 < d[2]) {
      d[2] = d2;   id[2] = i;
    }
  }
  for (int j = 0; j < k; ++j) {
    idx[(size_t)t * k + j] = id[j];
    w[(size_t)t * k + j] = 1.f / fmaxf(d[j], 1e-16f);
  }
}

// ---------------------------------------------------------------------------
// Build FP-level GEMM input: A[row, 0:csrc) = knn-interp(x_src),
// A[row, csrc:csrc+cskip) = skip features, rest zero padding.
// ---------------------------------------------------------------------------
__global__ void build_fp_A_kernel(const float* __restrict__ x_src, int c_src,
                                  int ns, const int* __restrict__ idx,
                                  const float* __restrict__ w, int k,
                                  const float* __restrict__ x_skip, int c_skip,
                                  float* __restrict__ A, int lda, int nt,
                                  int total) {
  int g = blockIdx.x * blockDim.x + threadIdx.x;
  if (g >= total) return;
  int col = g % lda;
  int rowg = g / lda;   // b*nt + t
  int b = rowg / nt;
  float val;
  if (col < c_src) {
    const float* xb = x_src + (size_t)b * ns * c_src;
    float num = 0.f, den = 0.f;
    for (int j = 0; j < k; ++j) {
      float ww = w[(size_t)rowg * k + j];
      den += ww;
      num += ww * xb[(size_t)idx[(size_t)rowg * k + j] * c_src + col];
    }
    val = num / den;
  } else if (col < c_src + c_skip) {
    val = x_skip[(size_t)rowg * c_skip + (col - c_src)];
  } else {
    val = 0.f;
  }
  A[g] = val;
}

// ===========================================================================
// Host-side orchestration
// ===========================================================================
extern "C" void kernel_launch(void* const* d_in, const int* in_sizes, int n_in,
                              void* d_out, int out_size, void* d_ws,
                              size_t ws_size, hipStream_t stream) {
  const int Bc = 8, K = 32;
  const float* x0 = (const float*)d_in[0];    // [8,4096,3]
  const float* pos0 = (const float*)d_in[1];  // [8,4096,3]

  // Weights: sa1..sa4 (3 layers each), fp4,fp3,fp2,fp1 (2 layers each)
  const float* Wt[20]; const float* Bt[20];
  {
    int p = 2;
    for (int i = 0; i < 20; ++i) {
      Wt[i] = (const float*)d_in[p++];
      Bt[i] = (const float*)d_in[p++];
    }
  }
  static const int layerCin[20]  = {6, 32, 32,  67, 64, 64,  131, 128, 128,
                                    259, 256, 256, 768, 256, 384, 256,
                                    320, 256, 131, 128};
  static const int layerCout[20] = {32, 32, 64,  64, 64, 128, 128, 128, 256,
                                    256, 256, 512, 256, 256, 256, 256,
                                    256, 128, 128, 256};

  // ---- workspace carve (bump allocator, 256B aligned) ----
  char* wsp = (char*)d_ws;
  auto alloc = [&](size_t bytes) -> void* {
    void* ptr = (void*)wsp;
    wsp += (bytes + 255) & ~((size_t)255);
    return ptr;
  };

  const int nlv[5] = {4096, 2048, 1024, 512, 256};
  const int cLv[5] = {3, 64, 128, 256, 512};
  const float r2a[4] = {0.0025f, 0.01f, 0.04f, 0.16f};
  const int c1a[4] = {32, 64, 128, 256};
  const int c2a[4] = {32, 64, 128, 256};
  const int c3a[4] = {64, 128, 256, 512};

  const float* posLv[5]; posLv[0] = pos0;
  float* xLv[5]; xLv[0] = nullptr;
  for (int l = 1; l <= 4; ++l) {
    posLv[l] = (float*)alloc((size_t)Bc * nlv[l] * 3 * 4);
    xLv[l] = (float*)alloc((size_t)Bc * nlv[l] * cLv[l] * 4);
  }
  int* idxL[4]; int* nbrL[4]; int* cntL[4];
  for (int l = 0; l < 4; ++l) {
    idxL[l] = (int*)alloc((size_t)Bc * nlv[l + 1] * 4);
    nbrL[l] = (int*)alloc((size_t)Bc * nlv[l + 1] * K * 4);
    cntL[l] = (int*)alloc((size_t)Bc * nlv[l + 1] * 4);
  }
  float* Abuf = (float*)alloc((size_t)Bc * 2048 * 320 * 4);  // max FP input
  float* Hbuf = (float*)alloc((size_t)16384 * 256 * 4);      // max FP hidden
  float* dec3 = (float*)alloc((size_t)Bc * 512 * 256 * 4);
  float* dec2 = (float*)alloc((size_t)Bc * 1024 * 256 * 4);
  float* dec1 = (float*)alloc((size_t)Bc * 2048 * 128 * 4);
  int* knnIdx = (int*)alloc((size_t)Bc * 4096 * 3 * 4);
  float* knnW = (float*)alloc((size_t)Bc * 4096 * 3 * 4);

  // ---- pad + pack all weights (K rounded to mult. of 4, K-pairs packed) ----
  float* Wp[20];
  for (int i = 0; i < 20; ++i) {
    const int ldk = (layerCin[i] + 3) & ~3;
    const int total = ldk * layerCout[i];
    Wp[i] = (float*)alloc((size_t)total * 4);
    pad_pack_kernel<<<(total + 255) / 256, 256, 0, stream>>>(
        Wt[i], layerCin[i], layerCout[i], Wp[i], total);
  }

  // ---- encoder: 4 SA levels ----
  const float* xsrc = x0;
  for (int l = 0; l < 4; ++l) {
    const int n = nlv[l], m = nlv[l + 1];
    const int tot = Bc * m;
    const size_t fpsLds = (size_t)(n + 2 * 256) * 4;
    fps_kernel<<<Bc, 256, fpsLds, stream>>>(posLv[l], n, m, idxL[l]);
    gather_pos_kernel<<<(tot + 255) / 256, 256, 0, stream>>>(
        posLv[l], n, idxL[l], m, (float*)posLv[l + 1], tot);
    radius_kernel<<<(tot + 255) / 256, 256, 0, stream>>>(
        posLv[l], n, posLv[l + 1], m, r2a[l], nbrL[l], cntL[l], tot);
    const int li = l * 3;
    const int lda1 = (cLv[l] + 6) & ~3;
    const size_t saLds =
        (size_t)(32 * lda1 + 32 * (c1a[l] + c2a[l]) + 128) * 4;
    sa_conv_kernel<<<tot, 128, saLds, stream>>>(
        xsrc, cLv[l], n, posLv[l], posLv[l + 1], nbrL[l], cntL[l],
        Wp[li], Bt[li], c1a[l], Wp[li + 1], Bt[li + 1], c2a[l],
        Wp[li + 2], Bt[li + 2], c3a[l], xLv[l + 1], m);
    xsrc = xLv[l + 1];
  }

  // ---- decoder: 4 FP levels ----
  auto run_fp = [&](const float* xs, int csrc, int ns, const float* psrc,
                    const float* ptgt, int nt, int kk, const float* xskip,
                    int cskip, const float* W1p, const float* B1, int cmid,
                    const float* W2p, const float* B2, int cout, float* out) {
    const int tt = Bc * nt;
    knn_kernel<<<(tt + 255) / 256, 256, 0, stream>>>(psrc, ns, ptgt, nt, kk,
                                                     knnIdx, knnW, tt);
    const int cin = csrc + cskip;
    const int lda = (cin + 3) & ~3;
    const int total = tt * lda;
    build_fp_A_kernel<<<(total + 255) / 256, 256, 0, stream>>>(
        xs, csrc, ns, knnIdx, knnW, kk, xskip, cskip, Abuf, lda, nt, total);
    dim3 g1(tt / 16, (cmid + 63) / 64);
    gemm_kernel<<<g1, 32, 0, stream>>>(Abuf, lda, W1p, cmid, B1, Hbuf, cmid, 1);
    dim3 g2(tt / 16, (cout + 63) / 64);
    gemm_kernel<<<g2, 32, 0, stream>>>(Hbuf, cmid, W2p, cout, B2, out, cout, 0);
  };

  // fp4: lvl4(256,512ch) -> lvl3(512), k=1, skip x3(256ch): 768->256->256
  run_fp(xLv[4], 512, 256, posLv[4], posLv[3], 512, 1, xLv[3], 256,
         Wp[12], Bt[12], 256, Wp[13], Bt[13], 256, dec3);
  // fp3: dec3(512,256ch) -> lvl2(1024), k=3, skip x2(128ch): 384->256->256
  run_fp(dec3, 256, 512, posLv[3], posLv[2], 1024, 3, xLv[2], 128,
         Wp[14], Bt[14], 256, Wp[15], Bt[15], 256, dec2);
  // fp2: dec2(1024,256ch) -> lvl1(2048), k=3, skip x1(64ch): 320->256->128
  run_fp(dec2, 256, 1024, posLv[2], posLv[1], 2048, 3, xLv[1], 64,
         Wp[16], Bt[16], 256, Wp[17], Bt[17], 128, dec1);
  // fp1: dec1(2048,128ch) -> lvl0(4096), k=3, skip x0(3ch): 131->128->256
  run_fp(dec1, 128, 2048, posLv[1], pos0, 4096, 3, x0, 3,
         Wp[18], Bt[18], 128, Wp[19], Bt[19], 256, (float*)d_out);
}